// CJSLoss_79130477461938
// MI455X (gfx1250) — compile-verified
//
#include <hip/hip_runtime.h>

// CJS weighted-reduction for MI455X (gfx1250, wave32).
// Memory-bound: 256 MiB stream @ 23.3 TB/s => ~11.5us floor.
// 3x v_log_f32 per element (co-limiting) + v_wmma_f32_16x16x4_f32 wave reduce.

typedef float f32x4 __attribute__((ext_vector_type(4)));
typedef float v2f   __attribute__((ext_vector_type(2)));
typedef float v8f   __attribute__((ext_vector_type(8)));

// Wave32 all-lane sum using one WMMA.
// A (16x4 f32, 2 VGPRs): lane L<16 holds A[L,0]=p (VGPR0), A[L,1]=0 (VGPR1);
//                        lane L>=16 holds A[L-16,2]=p, A[L-16,3]=0.
// B = all ones => D[m,n] = p[m] + p[m+16] (independent of n).
// D layout: VGPR r, lanes 0-15 = D[r,n] = p[r]+p[r+16]; lanes 16-31 = p[r+8]+p[r+24].
// So s = sum(d[0..7]) covers p[0..7]+p[16..23] on lanes 0-15 and
// p[8..15]+p[24..31] on lanes 16-31; s + shfl_xor(s,16) = full wave sum.
__device__ __forceinline__ float wave_sum_wmma(float p) {
    v2f a; a.x = p;   a.y = 0.0f;
    v2f b; b.x = 1.0f; b.y = 1.0f;
    v8f c = {};
    v8f d = __builtin_amdgcn_wmma_f32_16x16x4_f32(
        /*neg_a=*/false, a, /*neg_b=*/false, b,
        /*c_mod=*/(short)0, c, /*reuse_a=*/false, /*reuse_b=*/false);
    float s = ((d[0] + d[1]) + (d[2] + d[3])) + ((d[4] + d[5]) + (d[6] + d[7]));
    s += __shfl_xor(s, 16, 32);
    return s;
}

__global__ __launch_bounds__(256) void cjs_partial_kernel(
    const float* __restrict__ D, const float* __restrict__ P,
    float* __restrict__ ws, long long n4, unsigned cmask /* = C-1, C pow2 */) {
    const f32x4* D4 = (const f32x4*)D;
    const f32x4* P4 = (const f32x4*)P;

    const long long tid    = (long long)blockIdx.x * blockDim.x + threadIdx.x;
    const long long stride = (long long)gridDim.x * blockDim.x;

    const float LN2 = 0.69314718055994530942f;
    const float EPS = 1e-8f;
    const float Cf  = (float)(cmask + 1u);

    float acc = 0.0f;
    for (long long i = tid; i < n4; i += stride) {
        // Streaming, zero-reuse: non-temporal b128 loads (don't thrash 192MB L2).
        f32x4 dv = __builtin_nontemporal_load(D4 + i);
        f32x4 pv = __builtin_nontemporal_load(P4 + i);
        unsigned j0 = ((unsigned)(i << 2)) & cmask;  // column of element 0
        float wbase = Cf - (float)j0;                // weight (C - j)
#pragma unroll
        for (int k = 0; k < 4; ++k) {
            float dk = dv[k];
            float pk = pv[k];
            float s2 = dk + pk;                       // D + Dp
            // log m = log(0.5*(D+Dp+EPS)) = log(D+Dp+EPS) - ln2
            float lm = __logf(s2 + EPS) - LN2;
            float e  = fmaf(dk, __logf(dk),
                       fmaf(pk, __logf(pk), -s2 * lm));
            acc = fmaf(0.5f * (wbase - (float)k), e, acc);
        }
    }

    // Wave reduction via V_WMMA_F32_16X16X4_F32 (EXEC all-1s here: no divergence).
    float s = wave_sum_wmma(acc);

    __shared__ float smem[8];  // 256 threads = 8 wave32s
    const unsigned lane = threadIdx.x & 31u;
    const unsigned wave = threadIdx.x >> 5;
    if (lane == 0) smem[wave] = s;
    __syncthreads();
    if (threadIdx.x == 0) {
        float t = 0.0f;
        const int nw = blockDim.x >> 5;
        for (int w = 0; w < nw; ++w) t += smem[w];
        ws[blockIdx.x] = t;  // one deterministic partial per block
    }
}

__global__ __launch_bounds__(256) void cjs_final_kernel(
    const float* __restrict__ ws, float* __restrict__ out,
    int nblocks, double invB) {
    __shared__ double smem[256];
    double acc = 0.0;
    for (int i = threadIdx.x; i < nblocks; i += 256) acc += (double)ws[i];
    smem[threadIdx.x] = acc;
    __syncthreads();
    for (int off = 128; off > 0; off >>= 1) {
        if ((int)threadIdx.x < off) smem[threadIdx.x] += smem[threadIdx.x + off];
        __syncthreads();
    }
    if (threadIdx.x == 0) out[0] = (float)(smem[0] * invB);
}

extern "C" void kernel_launch(void* const* d_in, const int* in_sizes, int n_in,
                              void* d_out, int out_size, void* d_ws, size_t ws_size,
                              hipStream_t stream) {
    const float* D = (const float*)d_in[0];
    const float* P = (const float*)d_in[1];
    float* out = (float*)d_out;
    float* ws  = (float*)d_ws;

    const long long n  = (long long)in_sizes[0];  // B*C = 33,554,432
    const int       C  = 2048;                    // columns (power of two)
    const long long Bn = n / C;                   // rows
    const long long n4 = n >> 2;                  // float4 count (n % 4 == 0)

    int blocks = 4096;                            // ~8 float4 / thread, 8 waves/block
    long long maxb = (long long)(ws_size / sizeof(float));
    if (maxb < blocks) blocks = (int)maxb;
    if (blocks < 1) blocks = 1;

    cjs_partial_kernel<<<blocks, 256, 0, stream>>>(D, P, ws, n4, (unsigned)(C - 1));
    cjs_final_kernel<<<1, 256, 0, stream>>>(ws, out, blocks, 1.0 / (double)Bn);
}